// Seq2Seq_31696858645048
// MI455X (gfx1250) — compile-verified
//
#include <hip/hip_runtime.h>

// MI455X / gfx1250, wave32. One wave owns 16 batch rows; all LSTM gate
// matmuls are V_WMMA_F32_16X16X4_F32. State (h,c) stays in registers /
// per-wave LDS for the whole 256-step encoder scan + 11-step decoder.
// Activation pre-scales are folded into the weights; activations use the
// gfx1250 hardware v_tanh_f32. All gate combining is branchless
// (v_cndmask + ds_bpermute), only the o-lane h compute + LDS store is exec-masked.

typedef float v2f __attribute__((ext_vector_type(2)));
typedef float v4f __attribute__((ext_vector_type(4)));
typedef float v8f __attribute__((ext_vector_type(8)));

#define WAVES_PER_BLOCK 4
constexpr int S = 256, D = 8, T = 12;

#if __has_builtin(__builtin_amdgcn_tanhf)
__device__ __forceinline__ float fast_tanh(float x) {
  return __builtin_amdgcn_tanhf(x);   // v_tanh_f32
}
#else
__device__ __forceinline__ float fast_tanh(float x) {
  float s = __builtin_amdgcn_rcpf(1.0f + __expf(-2.0f * x));
  return fmaf(2.0f, s, -1.0f);
}
#endif

__device__ __forceinline__ float bperm(int byteIdx, float v) {
  return __int_as_float(
      __builtin_amdgcn_ds_bpermute(byteIdx, __float_as_int(v)));
}

__device__ __forceinline__ v8f wmma4(v2f a, v2f b, v8f c) {
  // D = A(16x4,f32) * B(4x16,f32) + C(16x16,f32)
  return __builtin_amdgcn_wmma_f32_16x16x4_f32(
      /*neg_a=*/false, a, /*neg_b=*/false, b,
      /*c_mod=*/(short)0, c, /*reuse_a=*/false, /*reuse_b=*/false);
}

struct LaneCtx {
  int n;        // column 0..15 (gate index)
  int half;     // lane >= 16
  int j2;       // K base for A/B layout: half*2
  int gt;       // gate type: 0=i 1=f 2=g 3=o
  int idx8;     // bpermute byte index for xor-8
  int idx4;     // bpermute byte index for xor-4
  bool selTP;   // gt==3  : first factor comes from partner (f)
  bool selC;    // gt odd : second factor is c (f/o lanes)
  float cS;     // column pre-scale folded into weights: g ? 1 : 0.5
  float cA, cB; // act = fma(tanh(acc), cA, cB)
};

// B-operand for W (row-major [16][K]), covering K rows kb..kb+3, pre-scaled
// by the lane's column scale (sigmoid cols absorb the /2 of
// sigmoid(z) = 0.5*tanh(z/2)+0.5 into the weights).
// B layout (4x16): VGPR0 = row (kb+j2), VGPR1 = row (kb+j2+1), col = n.
__device__ __forceinline__ v2f loadB(const float* W, int K, int kb,
                                     const LaneCtx& L) {
  v2f r;
  r.x = W[L.n * K + kb + L.j2] * L.cS;
  r.y = W[L.n * K + kb + L.j2 + 1] * L.cS;
  return r;
}

// LSTM cell elementwise update. acc = (pre-scaled) gates in D layout.
// c = cell state in replicated layout: every lane holds c[m][n&3].
// Produces h in A-matrix layout (via per-wave LDS transpose) into hA.
// hbuf layout: hbuf[j*16 + m] (64 floats per wave).
__device__ __forceinline__ void cell_update(v8f acc, v8f& c, float* hbuf,
                                            const LaneCtx& L, v2f& hA) {
  float act[8], tp[8], p[8];
#pragma unroll
  for (int v = 0; v < 8; ++v) {
    float t = fast_tanh(acc[v]);           // pre-scale already in weights
    act[v] = fmaf(t, L.cA, L.cB);          // sigmoid or tanh
  }
#pragma unroll
  for (int v = 0; v < 8; ++v) tp[v] = bperm(L.idx8, act[v]); // i<->g, f<->o
#pragma unroll
  for (int v = 0; v < 8; ++v) {
    float a1 = L.selTP ? tp[v] : act[v];   // o-lanes use f      (v_cndmask)
    float a2 = L.selC ? c[v] : tp[v];      // f/o lanes: *c      (v_cndmask)
    p[v] = a1 * a2;                        // i*g or f*c
  }
#pragma unroll
  for (int v = 0; v < 8; ++v)
    c[v] = p[v] + bperm(L.idx4, p[v]);     // c_new = f*c + i*g (replicated)
  // h = o * tanh(c_new); o lives as 'act' on gt==3 lanes. Transpose via LDS
  // using [j][m] layout -> two aligned b128 stores per o-lane.
  if (L.selTP) {
    float hv[8];
#pragma unroll
    for (int v = 0; v < 8; ++v) hv[v] = act[v] * fast_tanh(c[v]);
    float* dst = &hbuf[(L.n & 3) * 16 + L.half * 8];
    *(v4f*)dst       = (v4f){hv[0], hv[1], hv[2], hv[3]};
    *(v4f*)(dst + 4) = (v4f){hv[4], hv[5], hv[6], hv[7]};
  }
  __builtin_amdgcn_wave_barrier();   // same-wave DS is in-order; pin compiler
  hA.x = hbuf[L.j2 * 16 + L.n];          // h[m=n][j2]
  hA.y = hbuf[(L.j2 + 1) * 16 + L.n];    // h[m=n][j2+1]  (ds_load_2addr)
  __builtin_amdgcn_wave_barrier();
}

__global__ void __launch_bounds__(WAVES_PER_BLOCK * 32)
seq2seq_kernel(const float* __restrict__ src, const float* __restrict__ trg,
               const float* eW0, const float* eU0, const float* eb0a,
               const float* eb0b, const float* eW1, const float* eU1,
               const float* eb1a, const float* eb1b, const float* dW0,
               const float* dU0, const float* db0a, const float* db0b,
               const float* dW1, const float* dU1, const float* db1a,
               const float* db1b, const float* fcW, const float* fcb,
               float* __restrict__ out) {
  __shared__ __align__(16) float hbuf_all[WAVES_PER_BLOCK][64];
  const int lane = threadIdx.x & 31;
  const int wid = threadIdx.x >> 5;
  float* hbuf = hbuf_all[wid];

  LaneCtx L;
  L.n = lane & 15;
  L.half = lane >> 4;
  L.j2 = L.half * 2;
  L.gt = L.n >> 2;
  L.idx8 = (lane ^ 8) << 2;
  L.idx4 = (lane ^ 4) << 2;
  L.selTP = (L.gt == 3);
  L.selC = (L.gt & 1) != 0;
  const bool isG = (L.gt == 2);
  L.cS = isG ? 1.0f : 0.5f;
  L.cA = isG ? 1.0f : 0.5f;
  L.cB = isG ? 0.0f : 0.5f;

  const long bbase = ((long)blockIdx.x * WAVES_PER_BLOCK + wid) * 16;

  // --- encoder weights as WMMA B-operands (registers, loaded once) ---
  v2f eB0lo = loadB(eW0, 8, 0, L);   // Wih0 K=0..3
  v2f eB0hi = loadB(eW0, 8, 4, L);   // Wih0 K=4..7
  v2f eBU0  = loadB(eU0, 4, 0, L);   // Whh0
  v2f eBW1  = loadB(eW1, 4, 0, L);   // Wih1
  v2f eBU1  = loadB(eU1, 4, 0, L);   // Whh1
  const float biasE0 = (eb0a[L.n] + eb0b[L.n]) * L.cS;
  const float biasE1 = (eb1a[L.n] + eb1b[L.n]) * L.cS;
  v8f biasV0, biasV1;
#pragma unroll
  for (int v = 0; v < 8; ++v) { biasV0[v] = biasE0; biasV1[v] = biasE1; }

  // --- state ---
  v8f c0 = {}; v8f c1 = {};
  v2f h0A = {}; v2f h1A = {};

  const float* xp = src + (bbase + L.n) * (long)(S * D);

  // ===================== encoder scan =====================
  for (int s = 0; s < S; ++s) {
    if ((s & 7) == 0)
      __builtin_prefetch(xp + (s + 16) * D, 0, 1);  // global_prefetch_b8
    v2f xlo = *(const v2f*)(xp + s * D + L.j2);       // A op, K=0..3
    v2f xhi = *(const v2f*)(xp + s * D + 4 + L.j2);   // A op, K=4..7

    v8f acc = wmma4(xlo, eB0lo, biasV0);
    acc = wmma4(xhi, eB0hi, acc);
    acc = wmma4(h0A, eBU0, acc);
    cell_update(acc, c0, hbuf, L, h0A);

    acc = wmma4(h0A, eBW1, biasV1);
    acc = wmma4(h1A, eBU1, acc);
    cell_update(acc, c1, hbuf, L, h1A);
  }

  // ===================== decoder =====================
  v2f dBU0 = loadB(dU0, 4, 0, L);
  v2f dBW1 = loadB(dW1, 4, 0, L);
  v2f dBU1 = loadB(dU1, 4, 0, L);
  const float biasD0 = (db0a[L.n] + db0b[L.n]) * L.cS;
  const float biasD1 = (db1a[L.n] + db1b[L.n]) * L.cS;
  const float wIn = dW0[L.n] * L.cS;   // dec_Wih0 is (16,1)
  const float fw0 = fcW[L.j2], fw1 = fcW[L.j2 + 1];
  const float fb = fcb[0];
  v8f biasVD1;
#pragma unroll
  for (int v = 0; v < 8; ++v) biasVD1[v] = biasD1;

  // x0 = trg[:,0,:]; each lane keeps x[m] for its 8 rows (D-row layout)
  float xv[8];
#pragma unroll
  for (int v = 0; v < 8; ++v)
    xv[v] = trg[(bbase + v + L.half * 8) * T];
  if (L.half == 0) out[(bbase + L.n) * T + 0] = trg[(bbase + L.n) * T];

  for (int t = 1; t < T; ++t) {
    v8f acc;
#pragma unroll
    for (int v = 0; v < 8; ++v) acc[v] = fmaf(xv[v], wIn, biasD0); // x outer-prod
    acc = wmma4(h0A, dBU0, acc);
    cell_update(acc, c0, hbuf, L, h0A);

    acc = wmma4(h0A, dBW1, biasVD1);
    acc = wmma4(h1A, dBU1, acc);
    cell_update(acc, c1, hbuf, L, h1A);

    // pred[m] = h1[m]·fcW + fcb : partial dot per half, xor16 reduce
    float pp = h1A.x * fw0 + h1A.y * fw1;
    float pred = pp + __shfl_xor(pp, 16, 32) + fb;
    if (L.half == 0) out[(bbase + L.n) * T + t] = pred;
#pragma unroll
    for (int v = 0; v < 8; ++v)
      xv[v] = __shfl(pred, v + L.half * 8, 32);   // feed back as next x
  }
}

extern "C" void kernel_launch(void* const* d_in, const int* in_sizes, int n_in,
                              void* d_out, int out_size, void* d_ws,
                              size_t ws_size, hipStream_t stream) {
  const float* src  = (const float*)d_in[0];
  const float* trg  = (const float*)d_in[1];
  const float* eW0  = (const float*)d_in[2];
  const float* eU0  = (const float*)d_in[3];
  const float* eb0a = (const float*)d_in[4];
  const float* eb0b = (const float*)d_in[5];
  const float* eW1  = (const float*)d_in[6];
  const float* eU1  = (const float*)d_in[7];
  const float* eb1a = (const float*)d_in[8];
  const float* eb1b = (const float*)d_in[9];
  const float* dW0  = (const float*)d_in[10];
  const float* dU0  = (const float*)d_in[11];
  const float* db0a = (const float*)d_in[12];
  const float* db0b = (const float*)d_in[13];
  const float* dW1  = (const float*)d_in[14];
  const float* dU1  = (const float*)d_in[15];
  const float* db1a = (const float*)d_in[16];
  const float* db1b = (const float*)d_in[17];
  const float* fcW  = (const float*)d_in[18];
  const float* fcb  = (const float*)d_in[19];
  float* out = (float*)d_out;

  const int B = in_sizes[1] / T;                 // trg is (B, T, 1)
  const int blocks = B / (WAVES_PER_BLOCK * 16); // 16 rows per wave
  seq2seq_kernel<<<blocks, WAVES_PER_BLOCK * 32, 0, stream>>>(
      src, trg, eW0, eU0, eb0a, eb0b, eW1, eU1, eb1a, eb1b, dW0, dU0, db0a,
      db0b, dW1, dU1, db1a, db1b, fcW, fcb, out);
}